// InverseLeakySoftplus_11081015624046
// MI455X (gfx1250) — compile-verified
//
#include <hip/hip_runtime.h>
#include <cstdint>

// ---------------------------------------------------------------------------
// InverseLeakySoftplus via Newton, MI455X (gfx1250, wave32).
// Compute-bound on v_exp_f32/v_log_f32/v_rcp_f32; input streamed via CDNA5
// async global->LDS copies (ASYNCcnt), double-buffered per wave.
// ---------------------------------------------------------------------------

#define NEWTON_ITERS 10      // quadratic convergence
#define TPB          256     // 8 wave32 per workgroup
#define TILE_V4      256     // one float4 (16B) per thread per tile -> 4 KB tile
#define NBLOCKS      4096

#define AS1 __attribute__((address_space(1)))
#define AS3 __attribute__((address_space(3)))

typedef int v4i __attribute__((vector_size(4 * sizeof(int))));

// ---- fast reciprocal: single v_rcp_f32 (~1 ulp), no div_fixup expansion ---
__device__ __forceinline__ float fast_rcp(float x) {
  return __builtin_amdgcn_rcpf(x);
}

// ---- CDNA5 async global->LDS 128-bit copy (per-lane addresses) ------------
__device__ __forceinline__ void async_copy_b128(const float4* g, float4* l) {
#if __has_builtin(__builtin_amdgcn_global_load_async_to_lds_b128)
  __builtin_amdgcn_global_load_async_to_lds_b128(
      (AS1 v4i*)(uintptr_t)g,
      (AS3 v4i*)(uint32_t)(uintptr_t)l,   // low 32 bits of flat = LDS offset
      0 /*offset*/, 0 /*cpol*/);
#else
  uint32_t lds_off = (uint32_t)(uintptr_t)l;
  asm volatile("global_load_async_to_lds_b128 %0, %1, off"
               :: "v"(lds_off), "v"(g) : "memory");
#endif
}

// ---- s_wait_asynccnt <= N, plus compiler memory barrier -------------------
template <int N>
__device__ __forceinline__ void wait_async() {
#if __has_builtin(__builtin_amdgcn_s_wait_asynccnt)
  __builtin_amdgcn_s_wait_asynccnt(N);
#else
  asm volatile("s_wait_asynccnt %0" :: "n"(N) : "memory");
#endif
  asm volatile("" ::: "memory");   // keep ds_load from being hoisted above wait
}

// ---- Newton solve for f(x) = a*x + (1-a)*softplus(x) = y ------------------
// Per iteration: 1x v_exp_f32, 1x v_log_f32, 1x v_rcp_f32, ~12 other VALU.
__device__ __forceinline__ float newton_solve(float y, float a, float om_a,
                                              float inv_a) {
  float x = (y > 0.0f) ? y : y * inv_a;     // reference initializer
#pragma unroll
  for (int i = 0; i < NEWTON_ITERS; ++i) {
    float e   = __expf(-fabsf(x));                       // exp(-|x|)
    float sp  = __logf(1.0f + e) + fmaxf(x, 0.0f);       // stable softplus
    float fx  = fmaf(a, x, om_a * sp);                   // f(x)
    float opE = 1.0f + e;
    float u   = (x >= 0.0f) ? 1.0f : e;                  // sigmoid(x)*(1+e)
    float den = fmaf(a, opE, om_a * u);                  // f'(x)*(1+e)
    x -= (fx - y) * opE * fast_rcp(den);                 // single v_rcp_f32
  }
  return x;
}

__global__ __launch_bounds__(TPB)
void inv_leaky_softplus_kernel(const float* __restrict__ in,
                               const float* __restrict__ raw_alpha,
                               float* __restrict__ out, unsigned n) {
  __shared__ float4 tiles[2 * TILE_V4];    // 8 KB double buffer

  // alpha = 0.1 + 0.4*sigmoid(raw_alpha[0])  (stable for either sign)
  const float ra  = raw_alpha[0];
  const float er  = __expf(-fabsf(ra));
  const float rt  = fast_rcp(1.0f + er);
  const float sig = (ra >= 0.0f) ? rt : er * rt;
  const float a     = fmaf(0.4f, sig, 0.1f);
  const float om_a  = 1.0f - a;
  const float inv_a = fast_rcp(a);

  const unsigned tid    = threadIdx.x;
  const unsigned n_v4   = n >> 2;
  const unsigned ntiles = n_v4 / TILE_V4;
  const float4*  gin    = reinterpret_cast<const float4*>(in);
  float4*        gout   = reinterpret_cast<float4*>(out);

  // ---- async double-buffered pipeline over tiles --------------------------
  unsigned cur = 0;
  const unsigned t0 = blockIdx.x;
  if (t0 < ntiles)
    async_copy_b128(gin + (size_t)t0 * TILE_V4 + tid, &tiles[tid]);

  for (unsigned tl = t0; tl < ntiles; tl += gridDim.x) {
    const unsigned nxt = tl + gridDim.x;
    if (nxt < ntiles) {
      async_copy_b128(gin + (size_t)nxt * TILE_V4 + tid,
                      &tiles[(cur ^ 1) * TILE_V4 + tid]);
      wait_async<1>();   // current tile's copy has landed (in-order per wave)
    } else {
      wait_async<0>();
    }

    float4 v = tiles[cur * TILE_V4 + tid];   // ds_load_b128, own lane's data
    v.x = newton_solve(v.x, a, om_a, inv_a);
    v.y = newton_solve(v.y, a, om_a, inv_a);
    v.z = newton_solve(v.z, a, om_a, inv_a);
    v.w = newton_solve(v.w, a, om_a, inv_a);
    gout[(size_t)tl * TILE_V4 + tid] = v;    // global_store_b128

    cur ^= 1;
  }

  // ---- tail (n not a multiple of 4*TILE_V4; none for 8192x4096) -----------
  const unsigned done = ntiles * TILE_V4 * 4u;
  for (unsigned i = done + blockIdx.x * TPB + tid; i < n;
       i += gridDim.x * TPB) {
    out[i] = newton_solve(in[i], a, om_a, inv_a);
  }
}

extern "C" void kernel_launch(void* const* d_in, const int* in_sizes, int n_in,
                              void* d_out, int out_size, void* d_ws,
                              size_t ws_size, hipStream_t stream) {
  const float* in  = (const float*)d_in[0];
  const float* ra  = (const float*)d_in[1];
  float*       out = (float*)d_out;
  const unsigned n = (unsigned)in_sizes[0];
  inv_leaky_softplus_kernel<<<dim3(NBLOCKS), dim3(TPB), 0, stream>>>(in, ra,
                                                                     out, n);
}